// EMCapsule_71098888618589
// MI455X (gfx1250) — compile-verified
//
#include <hip/hip_runtime.h>
#include <math.h>

#define EPSV 1e-8f
#define Bn 64
#define On 64
#define In 1152
#define Dn 16
#define NCH 8
#define CH  (In / NCH)        // 144
#define PSTRIDE 528           // 16 (S) + 256 (E1) + 256 (E2)

typedef float v2f __attribute__((ext_vector_type(2)));
typedef float v8f __attribute__((ext_vector_type(8)));

// Sum across the 16 lanes of a half-wave (e dimension of the WMMA C/D tile).
__device__ __forceinline__ float hsum16(float x) {
    x += __shfl_xor(x, 1, 16);
    x += __shfl_xor(x, 2, 16);
    x += __shfl_xor(x, 4, 16);
    x += __shfl_xor(x, 8, 16);
    return x;
}

// Load the 4 K-block operand registers (A or B layout for V_WMMA_F32_16X16X4_F32).
__device__ __forceinline__ void load4(v2f (&dst)[4], const float* __restrict__ p, int hi2) {
#pragma unroll
    for (int kb = 0; kb < 4; ++kb) dst[kb] = *(const v2f*)(p + kb * 4 + hi2);
}

// Chained 16x16x16 fp32 matmul: four V_WMMA_F32_16X16X4_F32.
__device__ __forceinline__ v8f wmma16(const v2f (&a)[4], const v2f (&b)[4]) {
    v8f c = {};
#pragma unroll
    for (int kb = 0; kb < 4; ++kb)
        c = __builtin_amdgcn_wmma_f32_16x16x4_f32(false, a[kb], false, b[kb], (short)0, c, false, false);
    return c;
}

__device__ __forceinline__ void load8(float (&r)[8], const float* __restrict__ p) {
    float4 a = *(const float4*)(p);
    float4 b = *(const float4*)(p + 4);
    r[0]=a.x; r[1]=a.y; r[2]=a.z; r[3]=a.w; r[4]=b.x; r[5]=b.y; r[6]=b.z; r[7]=b.w;
}

// ---------------- Kernel 1: a_i = ||u + eps||, stored transposed (I,B) ----------------
__global__ void k_ai(const float* __restrict__ u, float* __restrict__ aiT) {
    int t = blockIdx.x * blockDim.x + threadIdx.x;   // t = b*In + i, exact grid
    int b = t / In, i = t % In;
    const float* p = u + (size_t)t * Dn;
    float s = 0.f;
#pragma unroll
    for (int d = 0; d < Dn; ++d) { float x = p[d] + EPSV; s += x * x; }
    aiT[(size_t)i * Bn + b] = sqrtf(s);
}

// ---------------- m-step iter 0, partial sums over an i-chunk ----------------
__device__ __forceinline__ void acc0(const v2f (&a)[4], const v2f (&b)[4], float bias_n,
                                     const float* __restrict__ arow,
                                     float (&S)[8], float (&E1)[8], float (&E2)[8]) {
    v8f c = wmma16(a, b);
    float ar[8]; load8(ar, arow);
#pragma unroll
    for (int v = 0; v < 8; ++v) {
        float r  = ar[v] * (1.0f / 64.0f);          // rr*a_i with rr = 1/O
        float Vv = c[v] + bias_n;
        S[v] += r; E1[v] += r * Vv; E2[v] += r * Vv * Vv;
    }
}

__global__ void k_mstep0_part(const float* __restrict__ u, const float* __restrict__ W,
                              const float* __restrict__ bias, const float* __restrict__ aiT,
                              float* __restrict__ part) {
    const int lane = threadIdx.x & 31;
    const int wv = threadIdx.x >> 5;
    const int o  = blockIdx.y * 4 + wv;
    const int bt = blockIdx.x, b0 = bt * 16;
    const int ch = blockIdx.z;
    const int m = lane & 15, hi = lane >> 4, hi2 = hi * 2;
    const float bias_n = bias[o * Dn + m] + EPSV;

    const float* uB = u + (size_t)(b0 + m) * In * Dn;
    const float* wB = W + (size_t)o * In * (Dn * Dn) + (size_t)m * Dn;
    const int i0 = ch * CH;

    float S[8], E1[8], E2[8];
#pragma unroll
    for (int v = 0; v < 8; ++v) { S[v] = 0.f; E1[v] = 0.f; E2[v] = 0.f; }

    v2f aP[4], bP[4], aQ[4], bQ[4];
    load4(aP, uB + (size_t)i0 * Dn, hi2);
    load4(bP, wB + (size_t)i0 * 256, hi2);

    for (int ii = 0; ii < CH; ii += 2) {
        const int ic = i0 + ii;
        const int ipf = (ic + 16 < In) ? ic + 16 : In - 1;
        __builtin_prefetch(wB + (size_t)ipf * 256, 0, 1);        // global_prefetch_b8
        load4(aQ, uB + (size_t)(ic + 1) * Dn, hi2);
        load4(bQ, wB + (size_t)(ic + 1) * 256, hi2);
        acc0(aP, bP, bias_n, aiT + (size_t)ic * Bn + b0 + hi * 8, S, E1, E2);
        const int inn = (ii + 2 < CH) ? ic + 2 : ic;             // clamped dummy on last pair
        load4(aP, uB + (size_t)inn * Dn, hi2);
        load4(bP, wB + (size_t)inn * 256, hi2);
        acc0(aQ, bQ, bias_n, aiT + (size_t)(ic + 1) * Bn + b0 + hi * 8, S, E1, E2);
    }

    const size_t base = (((size_t)bt * On + o) * NCH + ch) * PSTRIDE;
#pragma unroll
    for (int v = 0; v < 8; ++v) {
        const int row = v + hi * 8;
        const int re = row * 16 + m;
        if (m == 0) part[base + row] = S[v];
        part[base + 16 + re]  = E1[v];
        part[base + 272 + re] = E2[v];
    }
}

// ---------------- m-step iter 0, reduce + statistics ----------------
__global__ void k_mstep0_fin(const float* __restrict__ part,
                             const float* __restrict__ beta_a, const float* __restrict__ beta_u,
                             float* __restrict__ mean0, float* __restrict__ inv2v0,
                             float* __restrict__ ajp0) {
    const int lane = threadIdx.x & 31;
    const int wv = threadIdx.x >> 5;
    const int o  = blockIdx.y * 4 + wv;
    const int bt = blockIdx.x, b0 = bt * 16;
    const int m = lane & 15, hi = lane >> 4;

    float S[8], E1[8], E2[8];
#pragma unroll
    for (int v = 0; v < 8; ++v) { S[v] = 0.f; E1[v] = 0.f; E2[v] = 0.f; }
    const size_t base0 = ((size_t)bt * On + o) * NCH * PSTRIDE;
    for (int ch = 0; ch < NCH; ++ch) {
        const size_t base = base0 + (size_t)ch * PSTRIDE;
#pragma unroll
        for (int v = 0; v < 8; ++v) {
            const int row = v + hi * 8, re = row * 16 + m;
            S[v]  += part[base + row];
            E1[v] += part[base + 16 + re];
            E2[v] += part[base + 272 + re];
        }
    }
    const float bao = beta_a[o], buo = beta_u[o];
#pragma unroll
    for (int v = 0; v < 8; ++v) {
        const int row = b0 + v + hi * 8;
        float invS = 1.0f / (S[v] + EPSV);
        float mn   = E1[v] * invS;
        float vr   = (E2[v] - 2.0f * mn * E1[v] + mn * mn * S[v]) * invS + 1e-4f;
        float sl   = hsum16(__logf(vr));                     // sum_e log var
        float cost = S[v] * (16.0f * buo + sl);
        float aj   = 1.0f / (1.0f + __expf(-0.0005f * (bao - cost)));   // inv_temp iter0
        float p1   = sqrtf(6.2831853071795864f * __expf(sl) + EPSV);
        size_t idx = ((size_t)row * On + o) * Dn + m;
        mean0[idx]  = mn;
        inv2v0[idx] = 1.0f / (2.0f * vr + EPSV);
        if (m == 0) ajp0[(size_t)row * On + o] = aj / (p1 + EPSV);
    }
}

// ---------------- Kernel: e-step -> denom[b,i] = sum_o a_j * p_ij ----------------
__device__ __forceinline__ void estep_o(const v2f (&aK)[4], const v2f (&bK)[4], int o,
                                        int b0, int m, int hi,
                                        const float* __restrict__ bias,
                                        const float* __restrict__ mean0,
                                        const float* __restrict__ inv2v0,
                                        const float* __restrict__ ajp0, float (&dv)[8]) {
    v8f c = wmma16(aK, bK);
    const float bias_n = bias[o * Dn + m] + EPSV;
#pragma unroll
    for (int v = 0; v < 8; ++v) {
        const int row = b0 + v + hi * 8;
        size_t idx = ((size_t)row * On + o) * Dn + m;
        float Vv = c[v] + bias_n;
        float d  = Vv - mean0[idx];
        float s  = hsum16(d * d * inv2v0[idx]);              // sum_e
        dv[v] += ajp0[(size_t)row * On + o] * __expf(-s);
    }
}

__global__ void k_estep(const float* __restrict__ u, const float* __restrict__ W,
                        const float* __restrict__ bias,
                        const float* __restrict__ mean0, const float* __restrict__ inv2v0,
                        const float* __restrict__ ajp0, float* __restrict__ denomT) {
    const int lane = threadIdx.x & 31;
    const int wv = threadIdx.x >> 5;
    const int i  = blockIdx.y * 4 + wv;
    const int b0 = blockIdx.x * 16;
    const int m = lane & 15, hi = lane >> 4, hi2 = hi * 2;

    v2f aK[4];
    load4(aK, u + ((size_t)(b0 + m) * In + i) * Dn, hi2);

    float dv[8];
#pragma unroll
    for (int v = 0; v < 8; ++v) dv[v] = 0.f;

    v2f bP[4], bQ[4];
    load4(bP, W + (((size_t)0 * In + i) * Dn + m) * Dn, hi2);
    for (int o = 0; o < On; o += 2) {
        load4(bQ, W + (((size_t)(o + 1) * In + i) * Dn + m) * Dn, hi2);
        estep_o(aK, bP, o, b0, m, hi, bias, mean0, inv2v0, ajp0, dv);
        const int onn = (o + 2 < On) ? o + 2 : On - 1;       // clamped dummy on last pair
        load4(bP, W + (((size_t)onn * In + i) * Dn + m) * Dn, hi2);
        estep_o(aK, bQ, o + 1, b0, m, hi, bias, mean0, inv2v0, ajp0, dv);
    }
    if (m == 0) {
#pragma unroll
        for (int v = 0; v < 8; ++v)
            denomT[(size_t)i * Bn + b0 + v + hi * 8] = dv[v];
    }
}

// ---------------- m-step iter 1, partial sums over an i-chunk ----------------
__device__ __forceinline__ void acc1(const v2f (&a)[4], const v2f (&b)[4], float bias_n,
                                     const float* __restrict__ arow, const float* __restrict__ drow,
                                     const float (&mean_r)[8], const float (&tv2)[8], const float (&ajp)[8],
                                     float (&S)[8], float (&E1)[8], float (&E2)[8]) {
    v8f c = wmma16(a, b);
    float ar[8], dr[8];
    load8(ar, arow);
    load8(dr, drow);
#pragma unroll
    for (int v = 0; v < 8; ++v) {
        float Vv = c[v] + bias_n;
        float d  = Vv - mean_r[v];
        float s  = hsum16(d * d * tv2[v]);
        float ap = ajp[v] * __expf(-s);
        float rr = ap / (dr[v] + EPSV);
        float rra = rr * ar[v];
        S[v] += rra; E1[v] += rra * Vv; E2[v] += rra * Vv * Vv;
    }
}

__global__ void k_mstep1_part(const float* __restrict__ u, const float* __restrict__ W,
                              const float* __restrict__ bias, const float* __restrict__ aiT,
                              const float* __restrict__ mean0, const float* __restrict__ inv2v0,
                              const float* __restrict__ ajp0, const float* __restrict__ denomT,
                              float* __restrict__ part) {
    const int lane = threadIdx.x & 31;
    const int wv = threadIdx.x >> 5;
    const int o  = blockIdx.y * 4 + wv;
    const int bt = blockIdx.x, b0 = bt * 16;
    const int ch = blockIdx.z;
    const int m = lane & 15, hi = lane >> 4, hi2 = hi * 2;
    const float bias_n = bias[o * Dn + m] + EPSV;

    float mean_r[8], tv2[8], ajp[8];
#pragma unroll
    for (int v = 0; v < 8; ++v) {
        const int row = b0 + v + hi * 8;
        size_t idx = ((size_t)row * On + o) * Dn + m;
        mean_r[v] = mean0[idx];
        tv2[v]    = inv2v0[idx];
        ajp[v]    = ajp0[(size_t)row * On + o];
    }

    const float* uB = u + (size_t)(b0 + m) * In * Dn;
    const float* wB = W + (size_t)o * In * (Dn * Dn) + (size_t)m * Dn;
    const int i0 = ch * CH;

    float S[8], E1[8], E2[8];
#pragma unroll
    for (int v = 0; v < 8; ++v) { S[v] = 0.f; E1[v] = 0.f; E2[v] = 0.f; }

    v2f aP[4], bP[4], aQ[4], bQ[4];
    load4(aP, uB + (size_t)i0 * Dn, hi2);
    load4(bP, wB + (size_t)i0 * 256, hi2);

    for (int ii = 0; ii < CH; ii += 2) {
        const int ic = i0 + ii;
        const int ipf = (ic + 16 < In) ? ic + 16 : In - 1;
        __builtin_prefetch(wB + (size_t)ipf * 256, 0, 1);
        load4(aQ, uB + (size_t)(ic + 1) * Dn, hi2);
        load4(bQ, wB + (size_t)(ic + 1) * 256, hi2);
        acc1(aP, bP, bias_n, aiT + (size_t)ic * Bn + b0 + hi * 8,
             denomT + (size_t)ic * Bn + b0 + hi * 8, mean_r, tv2, ajp, S, E1, E2);
        const int inn = (ii + 2 < CH) ? ic + 2 : ic;
        load4(aP, uB + (size_t)inn * Dn, hi2);
        load4(bP, wB + (size_t)inn * 256, hi2);
        acc1(aQ, bQ, bias_n, aiT + (size_t)(ic + 1) * Bn + b0 + hi * 8,
             denomT + (size_t)(ic + 1) * Bn + b0 + hi * 8, mean_r, tv2, ajp, S, E1, E2);
    }

    const size_t base = (((size_t)bt * On + o) * NCH + ch) * PSTRIDE;
#pragma unroll
    for (int v = 0; v < 8; ++v) {
        const int row = v + hi * 8;
        const int re = row * 16 + m;
        if (m == 0) part[base + row] = S[v];
        part[base + 16 + re]  = E1[v];
        part[base + 272 + re] = E2[v];
    }
}

// ---------------- m-step iter 1, reduce + output ----------------
__global__ void k_mstep1_fin(const float* __restrict__ part,
                             const float* __restrict__ beta_a, const float* __restrict__ beta_u,
                             float* __restrict__ out) {
    const int lane = threadIdx.x & 31;
    const int wv = threadIdx.x >> 5;
    const int o  = blockIdx.y * 4 + wv;
    const int bt = blockIdx.x, b0 = bt * 16;
    const int m = lane & 15, hi = lane >> 4;

    float S[8], E1[8], E2[8];
#pragma unroll
    for (int v = 0; v < 8; ++v) { S[v] = 0.f; E1[v] = 0.f; E2[v] = 0.f; }
    const size_t base0 = ((size_t)bt * On + o) * NCH * PSTRIDE;
    for (int ch = 0; ch < NCH; ++ch) {
        const size_t base = base0 + (size_t)ch * PSTRIDE;
#pragma unroll
        for (int v = 0; v < 8; ++v) {
            const int row = v + hi * 8, re = row * 16 + m;
            S[v]  += part[base + row];
            E1[v] += part[base + 16 + re];
            E2[v] += part[base + 272 + re];
        }
    }
    const float bao = beta_a[o], buo = beta_u[o];
#pragma unroll
    for (int v = 0; v < 8; ++v) {
        const int row = b0 + v + hi * 8;
        float invS = 1.0f / (S[v] + EPSV);
        float mn   = E1[v] * invS;
        float vr   = (E2[v] - 2.0f * mn * E1[v] + mn * mn * S[v]) * invS + 1e-4f;
        float sl   = hsum16(__logf(vr));
        float cost = S[v] * (16.0f * buo + sl);
        float aj   = 1.0f / (1.0f + __expf(-0.000975f * (bao - cost)));  // inv_temp iter1
        float me   = mn + EPSV;
        float nm   = sqrtf(hsum16(me * me)) + EPSV;
        out[((size_t)row * On + o) * Dn + m] = aj * mn / nm;
    }
}

extern "C" void kernel_launch(void* const* d_in, const int* in_sizes, int n_in,
                              void* d_out, int out_size, void* d_ws, size_t ws_size,
                              hipStream_t stream) {
    const float* u      = (const float*)d_in[0];   // (B,I,16)
    const float* W      = (const float*)d_in[1];   // (1,O,I,16,16)
    const float* beta_a = (const float*)d_in[2];   // (1,O)
    const float* beta_u = (const float*)d_in[3];   // (1,O,1,1)
    const float* bias   = (const float*)d_in[4];   // (1,O,16)

    float* ws     = (float*)d_ws;
    float* aiT    = ws;                                   // I*B    = 73728
    float* mean0  = aiT    + (size_t)In * Bn;             // B*O*16 = 65536
    float* inv2v0 = mean0  + (size_t)Bn * On * Dn;        // 65536
    float* ajp0   = inv2v0 + (size_t)Bn * On * Dn;        // B*O    = 4096
    float* denomT = ajp0   + (size_t)Bn * On;             // I*B    = 73728
    float* part   = denomT + (size_t)In * Bn;             // 2048*528 = 1081344
    float* outp   = (float*)d_out;

    k_ai<<<dim3((Bn * In) / 256), 256, 0, stream>>>(u, aiT);
    k_mstep0_part<<<dim3(4, 16, NCH), 128, 0, stream>>>(u, W, bias, aiT, part);
    k_mstep0_fin<<<dim3(4, 16), 128, 0, stream>>>(part, beta_a, beta_u, mean0, inv2v0, ajp0);
    k_estep<<<dim3(4, In / 4), 128, 0, stream>>>(u, W, bias, mean0, inv2v0, ajp0, denomT);
    k_mstep1_part<<<dim3(4, 16, NCH), 128, 0, stream>>>(u, W, bias, aiT, mean0, inv2v0,
                                                        ajp0, denomT, part);
    k_mstep1_fin<<<dim3(4, 16), 128, 0, stream>>>(part, beta_a, beta_u, outp);
}